// PointNet2MSGPartSeg_15564961481178
// MI455X (gfx1250) — compile-verified
//
#include <hip/hip_runtime.h>
#include <hip/hip_bf16.h>

typedef float v2f __attribute__((ext_vector_type(2)));
typedef float v8f __attribute__((ext_vector_type(8)));

#define BN_EPS 1e-5f

// ============================================================
// GEMM via fp32 WMMA:  Y[b] = W (Co x Ci) * X[b] (Ci x M)
// X layout [B][Ci][M], Y layout [B][Co][M]. M must be mult of 16.
// Each wave: 4 Co-tiles x 1 N-tile (64x16 output), B-fragment reused 4x.
// Main K loop is branch-free; K remainder handled by one masked tail step.
// ============================================================
__global__ __launch_bounds__(256) void gemm_wmma_f32(
    const float* __restrict__ W, const float* __restrict__ X,
    float* __restrict__ Y, int Co, int Ci, int M)
{
  const int b = blockIdx.z;
  const float* Xb = X + (size_t)b * Ci * M;
  float*       Yb = Y + (size_t)b * Co * M;
  const int tilesN  = M >> 4;
  const int tilesCo = (Co + 15) >> 4;
  const int grpCo   = (tilesCo + 3) >> 2;        // groups of 4 Co-tiles
  const int wave = threadIdx.x >> 5;
  const int lane = threadIdx.x & 31;
  const int gidx = blockIdx.x * (blockDim.x >> 5) + wave;
  if (gidx >= grpCo * tilesN) return;
  const int gco = gidx / tilesN;
  const int tn  = gidx - gco * tilesN;
  const int n0  = tn << 4;
  const int l16 = lane & 15;
  const bool hi = lane >= 16;
  const int koff = hi ? 2 : 0;                   // lanes 16-31 hold K+2,K+3
  const int col  = n0 + l16;                     // B/D column (spatial)
  const int co0  = gco << 6;

  // Per-tile A row offsets, clamped to a valid row: rows >= Co load real
  // memory but their results are never stored.
  unsigned rowoff[4];
#pragma unroll
  for (int t = 0; t < 4; ++t) {
    int r = co0 + t * 16 + l16;
    if (r > Co - 1) r = Co - 1;
    rowoff[t] = (unsigned)r * (unsigned)Ci;
  }

  const float* Xc = Xb + col;
  const unsigned uM = (unsigned)M;

  v8f acc[4] = {};
  const int kmain = Ci & ~3;
  int k0 = 0;
  for (; k0 < kmain; k0 += 4) {
    const unsigned ka = (unsigned)(k0 + koff);
    v2f bv;
    bv.x = Xc[ka * uM];
    bv.y = Xc[(ka + 1u) * uM];
    v2f a[4];
#pragma unroll
    for (int t = 0; t < 4; ++t) {
      a[t].x = W[rowoff[t] + ka];
      a[t].y = W[rowoff[t] + ka + 1u];
    }
#pragma unroll
    for (int t = 0; t < 4; ++t)
      acc[t] = __builtin_amdgcn_wmma_f32_16x16x4_f32(
          false, a[t], false, bv, (short)0, acc[t], false, false);
  }
  if (k0 < Ci) {
    // K remainder (1..3): clamp addresses, zero the B operand beyond Ci.
    const int ka  = k0 + koff;
    const unsigned kc0 = (unsigned)(ka     < Ci ? ka     : Ci - 1);
    const unsigned kc1 = (unsigned)(ka + 1 < Ci ? ka + 1 : Ci - 1);
    const float bx = Xc[kc0 * uM];
    const float by = Xc[kc1 * uM];
    v2f bv;
    bv.x = (ka     < Ci) ? bx : 0.0f;
    bv.y = (ka + 1 < Ci) ? by : 0.0f;
    v2f a[4];
#pragma unroll
    for (int t = 0; t < 4; ++t) {
      a[t].x = W[rowoff[t] + kc0];
      a[t].y = W[rowoff[t] + kc1];
    }
#pragma unroll
    for (int t = 0; t < 4; ++t)
      acc[t] = __builtin_amdgcn_wmma_f32_16x16x4_f32(
          false, a[t], false, bv, (short)0, acc[t], false, false);
  }

#pragma unroll
  for (int t = 0; t < 4; ++t) {
#pragma unroll
    for (int j = 0; j < 8; ++j) {
      const int row = co0 + t * 16 + j + (hi ? 8 : 0);
      if (row < Co) Yb[(size_t)row * M + col] = acc[t][j];
    }
  }
}

// ============================================================
// BatchNorm (training-mode) statistics: one block per channel,
// deterministic tree reduction. stats[2c]=mean, stats[2c+1]=rsqrt(var+eps)
// ============================================================
__global__ __launch_bounds__(256) void bn_stats_kernel(
    const float* __restrict__ Y, int C, int B, int M, float* __restrict__ stats)
{
  const int c = blockIdx.x;
  __shared__ float s1[256], s2[256];
  float a1 = 0.f, a2 = 0.f;
  const int total = B * M;
  for (int i = threadIdx.x; i < total; i += blockDim.x) {
    const int b = i / M;
    const int m = i - b * M;
    const float v = Y[((size_t)b * C + c) * M + m];
    a1 += v; a2 += v * v;
  }
  s1[threadIdx.x] = a1; s2[threadIdx.x] = a2;
  __syncthreads();
  for (int off = 128; off > 0; off >>= 1) {
    if ((int)threadIdx.x < off) {
      s1[threadIdx.x] += s1[threadIdx.x + off];
      s2[threadIdx.x] += s2[threadIdx.x + off];
    }
    __syncthreads();
  }
  if (threadIdx.x == 0) {
    const float inv_n = 1.0f / (float)total;
    const float mean = s1[0] * inv_n;
    const float var  = s2[0] * inv_n - mean * mean;
    stats[2 * c]     = mean;
    stats[2 * c + 1] = rsqrtf(var + BN_EPS);
  }
}

__global__ __launch_bounds__(256) void bn_relu_kernel(
    float* __restrict__ Y, const float* __restrict__ g, const float* __restrict__ bp,
    const float* __restrict__ stats, int C, int M, size_t total)
{
  const size_t t = (size_t)blockIdx.x * blockDim.x + threadIdx.x;
  if (t >= total) return;
  const int c = (int)((t / (size_t)M) % (size_t)C);
  float v = Y[t];
  v = (v - stats[2 * c]) * stats[2 * c + 1];
  v = v * g[c] + bp[c];
  Y[t] = fmaxf(v, 0.f);
}

// ============================================================
// Farthest point sampling: one block per batch, sequential steps.
// ============================================================
__global__ __launch_bounds__(256) void fps_kernel(
    const float* __restrict__ xyz, int n, int m, int* __restrict__ idx)
{
  const int b = blockIdx.x;
  const float* px = xyz + (size_t)b * 3 * n;
  const float* py = px + n;
  const float* pz = py + n;
  extern __shared__ char smem_raw[];
  float* dmin = (float*)smem_raw;            // n floats
  float* rv   = dmin + n;                    // blockDim floats
  int*   ri   = (int*)(rv + blockDim.x);     // blockDim ints
  for (int i = threadIdx.x; i < n; i += blockDim.x) dmin[i] = 1e10f;
  if (threadIdx.x == 0) idx[(size_t)b * m] = 0;
  __syncthreads();
  int last = 0;
  for (int s = 1; s < m; ++s) {
    const float lx = px[last], ly = py[last], lz = pz[last];
    float best = -1.f; int bi = 0;
    for (int i = threadIdx.x; i < n; i += blockDim.x) {
      const float dx = px[i] - lx, dy = py[i] - ly, dz = pz[i] - lz;
      const float d = dx * dx + dy * dy + dz * dz;
      const float dm = fminf(dmin[i], d);
      dmin[i] = dm;
      if (dm > best) { best = dm; bi = i; }
    }
    rv[threadIdx.x] = best; ri[threadIdx.x] = bi;
    __syncthreads();
    for (int off = 128; off > 0; off >>= 1) {
      if ((int)threadIdx.x < off) {
        const float o = rv[threadIdx.x + off];
        const int oi = ri[threadIdx.x + off];
        if (o > rv[threadIdx.x] ||
            (o == rv[threadIdx.x] && oi < ri[threadIdx.x])) {
          rv[threadIdx.x] = o; ri[threadIdx.x] = oi;
        }
      }
      __syncthreads();
    }
    last = ri[0];
    if (threadIdx.x == 0) idx[(size_t)b * m + s] = last;
    __syncthreads();
  }
}

// gather columns: dst[b][c][j] = src[b][c][idx[b][j]]
__global__ __launch_bounds__(256) void gather_cols_kernel(
    const float* __restrict__ src, const int* __restrict__ idx,
    float* __restrict__ dst, int C, int n, int m, int B)
{
  const size_t t = (size_t)blockIdx.x * blockDim.x + threadIdx.x;
  const size_t total = (size_t)B * C * m;
  if (t >= total) return;
  const int j = (int)(t % m);
  const size_t bc = t / m;
  const int c = (int)(bc % C);
  const int b = (int)(bc / C);
  dst[t] = src[((size_t)b * C + c) * n + idx[(size_t)b * m + j]];
}

// ball query: first k in-order indices within radius, padded (ref semantics)
__global__ __launch_bounds__(256) void ball_query_kernel(
    const float* __restrict__ cx, const float* __restrict__ xyz,
    int S, int n, float r2, int k, int B, int* __restrict__ out)
{
  const int t = blockIdx.x * blockDim.x + threadIdx.x;
  if (t >= B * S) return;
  const int s = t % S, b = t / S;
  const float qx = cx[((size_t)b * 3 + 0) * S + s];
  const float qy = cx[((size_t)b * 3 + 1) * S + s];
  const float qz = cx[((size_t)b * 3 + 2) * S + s];
  const float* xp = xyz + (size_t)b * 3 * n;
  int* o = out + (size_t)t * k;
  int cnt = 0, first = -1;
  for (int i = 0; i < n && cnt < k; ++i) {
    const float dx = xp[i] - qx, dy = xp[n + i] - qy, dz = xp[2 * n + i] - qz;
    if (dx * dx + dy * dy + dz * dz < r2) {
      if (first < 0) first = i;
      o[cnt++] = i;
    }
  }
  const int pad = first < 0 ? 0 : first;
  for (; cnt < k; ++cnt) o[cnt] = pad;
}

// build grouped tensor g[b][c][s][j]: xyz-diff channels then gathered features
__global__ __launch_bounds__(256) void group_kernel(
    const float* __restrict__ xyz, const float* __restrict__ nxyz,
    const float* __restrict__ feat, const int* __restrict__ idx,
    int n, int S, int k, int Cf, int B, float* __restrict__ g)
{
  const size_t t = (size_t)blockIdx.x * blockDim.x + threadIdx.x;
  const size_t total = (size_t)B * (3 + Cf) * S * k;
  if (t >= total) return;
  const int j = (int)(t % k);
  size_t r = t / k;
  const int s = (int)(r % S); r /= S;
  const int c = (int)(r % (3 + Cf));
  const int b = (int)(r / (3 + Cf));
  const int id = idx[((size_t)b * S + s) * k + j];
  float v;
  if (c < 3)
    v = xyz[((size_t)b * 3 + c) * n + id] - nxyz[((size_t)b * 3 + c) * S + s];
  else
    v = feat[((size_t)b * Cf + (c - 3)) * n + id];
  g[t] = v;
}

// max over neighbor dim, writing into concat channel slot
__global__ __launch_bounds__(256) void maxpool_k_kernel(
    const float* __restrict__ in, float* __restrict__ out,
    int C, int S, int k, int Ct, int coff, int B)
{
  const int t = blockIdx.x * blockDim.x + threadIdx.x;
  if (t >= B * C * S) return;
  const int s = t % S;
  const int r = t / S;
  const int c = r % C;
  const int b = r / C;
  const float* p = in + (((size_t)b * C + c) * S + s) * k;
  float m = p[0];
  for (int j = 1; j < k; ++j) m = fmaxf(m, p[j]);
  out[((size_t)b * Ct + coff + c) * S + s] = m;
}

// copy channel block into concat buffer
__global__ __launch_bounds__(256) void copy_channels_kernel(
    const float* __restrict__ src, float* __restrict__ dst,
    int Cs, int off, int Ct, int M, int B)
{
  const size_t t = (size_t)blockIdx.x * blockDim.x + threadIdx.x;
  const size_t total = (size_t)B * Cs * M;
  if (t >= total) return;
  const int m = (int)(t % M);
  const size_t bc = t / M;
  const int c = (int)(bc % Cs);
  const int b = (int)(bc / Cs);
  dst[((size_t)b * Ct + off + c) * M + m] = src[t];
}

__global__ __launch_bounds__(256) void onehot_kernel(
    const int* __restrict__ cls, float* __restrict__ dst,
    int N, int Ct, int off, int K, int B)
{
  const size_t t = (size_t)blockIdx.x * blockDim.x + threadIdx.x;
  const size_t total = (size_t)B * K * N;
  if (t >= total) return;
  const int i = (int)(t % N);
  const size_t bc = t / N;
  const int j = (int)(bc % K);
  const int b = (int)(bc / K);
  dst[((size_t)b * Ct + off + j) * N + i] = (cls[b] == j) ? 1.0f : 0.0f;
}

// 3-nearest-neighbor + normalized inverse-distance weights
__global__ __launch_bounds__(256) void knn3_kernel(
    const float* __restrict__ q, const float* __restrict__ kx,
    int Nq, int Nk, int B, int* __restrict__ oidx, float* __restrict__ ow)
{
  const int t = blockIdx.x * blockDim.x + threadIdx.x;
  if (t >= B * Nq) return;
  const int b = t / Nq, i = t - b * Nq;
  const float qx = q[((size_t)b * 3 + 0) * Nq + i];
  const float qy = q[((size_t)b * 3 + 1) * Nq + i];
  const float qz = q[((size_t)b * 3 + 2) * Nq + i];
  const float* kp = kx + (size_t)b * 3 * Nk;
  float bd0 = 3.4e38f, bd1 = 3.4e38f, bd2 = 3.4e38f;
  int i0 = 0, i1 = 0, i2 = 0;
  for (int jj = 0; jj < Nk; ++jj) {
    const float dx = kp[jj] - qx, dy = kp[Nk + jj] - qy, dz = kp[2 * Nk + jj] - qz;
    const float d = dx * dx + dy * dy + dz * dz;
    if (d < bd0)      { bd2 = bd1; i2 = i1; bd1 = bd0; i1 = i0; bd0 = d; i0 = jj; }
    else if (d < bd1) { bd2 = bd1; i2 = i1; bd1 = d; i1 = jj; }
    else if (d < bd2) { bd2 = d; i2 = jj; }
  }
  float w0 = 1.f / (fmaxf(bd0, 0.f) + 1e-8f);
  float w1 = 1.f / (fmaxf(bd1, 0.f) + 1e-8f);
  float w2 = 1.f / (fmaxf(bd2, 0.f) + 1e-8f);
  const float s = w0 + w1 + w2;
  w0 /= s; w1 /= s; w2 /= s;
  int* op = oidx + (size_t)t * 3;
  float* wp = ow + (size_t)t * 3;
  op[0] = i0; op[1] = i1; op[2] = i2;
  wp[0] = w0; wp[1] = w1; wp[2] = w2;
}

// interpolated feature into concat buffer channels [0,Ck)
__global__ __launch_bounds__(256) void interp3_kernel(
    const float* __restrict__ kf, const int* __restrict__ idx,
    const float* __restrict__ w, float* __restrict__ dst,
    int Ck, int Nk, int Nq, int Ct, int B)
{
  const size_t t = (size_t)blockIdx.x * blockDim.x + threadIdx.x;
  const size_t total = (size_t)B * Ck * Nq;
  if (t >= total) return;
  const int qq = (int)(t % Nq);
  const size_t bc = t / Nq;
  const int c = (int)(bc % Ck);
  const int b = (int)(bc / Ck);
  const int* ip = idx + ((size_t)b * Nq + qq) * 3;
  const float* wp = w + ((size_t)b * Nq + qq) * 3;
  const float* f = kf + ((size_t)b * Ck + c) * Nk;
  dst[((size_t)b * Ct + c) * Nq + qq] =
      f[ip[0]] * wp[0] + f[ip[1]] * wp[1] + f[ip[2]] * wp[2];
}

__global__ __launch_bounds__(256) void bias_add_kernel(
    float* __restrict__ Y, const float* __restrict__ bias, int C, int M, size_t total)
{
  const size_t t = (size_t)blockIdx.x * blockDim.x + threadIdx.x;
  if (t >= total) return;
  const int c = (int)((t / (size_t)M) % (size_t)C);
  Y[t] += bias[c];
}

// ============================================================
// Host side
// ============================================================
struct Lyr { const float *b, *g, *w; int cin, cout; };

static inline unsigned nblk(size_t total, int t) {
  return (unsigned)((total + (size_t)t - 1) / (size_t)t);
}

static void launch_gemm(const float* W, const float* X, float* Y,
                        int Co, int Ci, int M, int B, hipStream_t st) {
  const int tilesCo = (Co + 15) / 16;
  const int grpCo   = (tilesCo + 3) / 4;
  const int tiles   = grpCo * (M / 16);
  dim3 grid((unsigned)((tiles + 7) / 8), 1, (unsigned)B);
  gemm_wmma_f32<<<grid, 256, 0, st>>>(W, X, Y, Co, Ci, M);
}

// shared MLP: per layer GEMM -> BN stats -> BN+ReLU (in place). Ping-pong.
static const float* run_mlp(const Lyr* L, int nl, const float* X, int B, int M,
                            float* bufA, float* bufB, float* stats, hipStream_t st) {
  const float* in = X;
  float* out = bufA;
  for (int i = 0; i < nl; ++i) {
    const int Co = L[i].cout, Ci = L[i].cin;
    launch_gemm(L[i].w, in, out, Co, Ci, M, B, st);
    bn_stats_kernel<<<(unsigned)Co, 256, 0, st>>>(out, Co, B, M, stats);
    const size_t tot = (size_t)B * Co * M;
    bn_relu_kernel<<<nblk(tot, 256), 256, 0, st>>>(out, L[i].g, L[i].b, stats, Co, M, tot);
    in = out;
    out = (out == bufA) ? bufB : bufA;
  }
  return in;
}

extern "C" void kernel_launch(void* const* d_in, const int* in_sizes, int n_in,
                              void* d_out, int out_size, void* d_ws, size_t ws_size,
                              hipStream_t stream) {
  (void)in_sizes; (void)n_in; (void)out_size; (void)ws_size;
  const int B = 8, N0 = 2048, S1 = 512, S2 = 128;

  const float* points = (const float*)d_in[0];  // [B][3][N0]
  const int*   cls    = (const int*)d_in[1];    // [B]

  // -------- parameter leaves (pytree order: dict keys sorted; layer dict -> b,g,w)
  int p = 2;
  auto L = [&](int cin, int cout) {
    Lyr l;
    l.b = (const float*)d_in[p++];
    l.g = (const float*)d_in[p++];
    l.w = (const float*)d_in[p++];
    l.cin = cin; l.cout = cout;
    return l;
  };
  Lyr fp0L[2] = { L(1536, 256), L(256, 256) };
  Lyr fp1L[2] = { L(576, 256),  L(256, 128) };
  Lyr fp2L[2] = { L(147, 128),  L(128, 128) };
  Lyr locL[3] = { L(515, 256),  L(256, 512), L(512, 1024) };
  Lyr segL[1] = { L(128, 128) };
  Lyr sa00[3] = { L(3, 32),   L(32, 32),   L(32, 64) };
  Lyr sa01[3] = { L(3, 64),   L(64, 64),   L(64, 128) };
  Lyr sa02[3] = { L(3, 64),   L(64, 96),   L(96, 128) };
  Lyr sa10[3] = { L(323, 128), L(128, 128), L(128, 256) };
  Lyr sa11[3] = { L(323, 128), L(128, 196), L(196, 256) };
  const float* seg_b = (const float*)d_in[p++];
  const float* seg_w = (const float*)d_in[p++];   // [50][128]

  // -------- workspace bump allocator
  char* wsp = (char*)d_ws;
  auto allocf = [&](size_t nelem) -> float* {
    float* r = (float*)wsp;
    wsp += ((nelem * 4) + 255) & ~(size_t)255;
    return r;
  };
  auto alloci = [&](size_t nelem) -> int* {
    int* r = (int*)wsp;
    wsp += ((nelem * 4) + 255) & ~(size_t)255;
    return r;
  };

  float* feat0 = allocf((size_t)B * 19 * N0);
  float* xyz1  = allocf((size_t)B * 3 * S1);
  float* xyz2  = allocf((size_t)B * 3 * S2);
  int*   fidx  = alloci((size_t)B * S1);
  int*   bidx  = alloci((size_t)B * S1 * 128);
  float* gbuf  = allocf((size_t)B * 323 * S2 * 128);   // max grouped input
  float* actA  = allocf((size_t)B * 128 * S1 * 128);   // max activation
  float* actB  = allocf((size_t)B * 128 * S1 * 128);
  float* stats = allocf(2 * 1024);
  float* sa1o  = allocf((size_t)B * 320 * S1);
  float* sa2o  = allocf((size_t)B * 512 * S2);
  float* locin = allocf((size_t)B * 515 * S2);
  int*   ki    = alloci((size_t)B * N0 * 3);
  float* kw    = allocf((size_t)B * N0 * 3);
  float* fpin  = allocf((size_t)B * 147 * N0);         // max concat input

  // -------- inter_feature[0] = [points ; one_hot]
  copy_channels_kernel<<<nblk((size_t)B * 3 * N0, 256), 256, 0, stream>>>(
      points, feat0, 3, 0, 19, N0, B);
  onehot_kernel<<<nblk((size_t)B * 16 * N0, 256), 256, 0, stream>>>(
      cls, feat0, N0, 19, 3, 16, B);

  // -------- SA layer 1 (N0 -> S1 centroids, 3 scales)
  fps_kernel<<<B, 256, (size_t)(N0 + 512) * 4, stream>>>(points, N0, S1, fidx);
  gather_cols_kernel<<<nblk((size_t)B * 3 * S1, 256), 256, 0, stream>>>(
      points, fidx, xyz1, 3, N0, S1, B);
  {
    const Lyr* Ls[3] = { sa00, sa01, sa02 };
    const float rs[3] = { 0.1f, 0.2f, 0.4f };
    const int ks[3] = { 32, 64, 128 };
    const int offs[3] = { 0, 64, 192 };
    const int couts[3] = { 64, 128, 128 };
    for (int sc = 0; sc < 3; ++sc) {
      const int k = ks[sc];
      const float r2 = rs[sc] * rs[sc];
      ball_query_kernel<<<nblk((size_t)B * S1, 256), 256, 0, stream>>>(
          xyz1, points, S1, N0, r2, k, B, bidx);
      const size_t gt = (size_t)B * 3 * S1 * k;
      group_kernel<<<nblk(gt, 256), 256, 0, stream>>>(
          points, xyz1, (const float*)nullptr, bidx, N0, S1, k, 0, B, gbuf);
      const float* o = run_mlp(Ls[sc], 3, gbuf, B, S1 * k, actA, actB, stats, stream);
      maxpool_k_kernel<<<nblk((size_t)B * couts[sc] * S1, 256), 256, 0, stream>>>(
          o, sa1o, couts[sc], S1, k, 320, offs[sc], B);
    }
  }

  // -------- SA layer 2 (S1 -> S2 centroids, 2 scales, feat 320ch)
  fps_kernel<<<B, 256, (size_t)(S1 + 512) * 4, stream>>>(xyz1, S1, S2, fidx);
  gather_cols_kernel<<<nblk((size_t)B * 3 * S2, 256), 256, 0, stream>>>(
      xyz1, fidx, xyz2, 3, S1, S2, B);
  {
    const Lyr* Ls[2] = { sa10, sa11 };
    const float rs[2] = { 0.4f, 0.8f };
    const int ks[2] = { 64, 128 };
    const int offs[2] = { 0, 256 };
    const int couts[2] = { 256, 256 };
    for (int sc = 0; sc < 2; ++sc) {
      const int k = ks[sc];
      const float r2 = rs[sc] * rs[sc];
      ball_query_kernel<<<nblk((size_t)B * S2, 256), 256, 0, stream>>>(
          xyz2, xyz1, S2, S1, r2, k, B, bidx);
      const size_t gt = (size_t)B * 323 * S2 * k;
      group_kernel<<<nblk(gt, 256), 256, 0, stream>>>(
          xyz1, xyz2, sa1o, bidx, S1, S2, k, 320, B, gbuf);
      const float* o = run_mlp(Ls[sc], 3, gbuf, B, S2 * k, actA, actB, stats, stream);
      maxpool_k_kernel<<<nblk((size_t)B * couts[sc] * S2, 256), 256, 0, stream>>>(
          o, sa2o, couts[sc], S2, k, 512, offs[sc], B);
    }
  }

  // -------- mlp_local: concat [xyz2 ; sa2o] (515ch @ S2)
  copy_channels_kernel<<<nblk((size_t)B * 3 * S2, 256), 256, 0, stream>>>(
      xyz2, locin, 3, 0, 515, S2, B);
  copy_channels_kernel<<<nblk((size_t)B * 512 * S2, 256), 256, 0, stream>>>(
      sa2o, locin, 512, 3, 515, S2, B);
  const float* mlo = run_mlp(locL, 3, locin, B, S2, actA, actB, stats, stream);  // 1024ch

  // -------- FP stage 0: query=xyz2(S2), key=xyz2(S2), 1024+512 -> (256,256)
  knn3_kernel<<<nblk((size_t)B * S2, 256), 256, 0, stream>>>(xyz2, xyz2, S2, S2, B, ki, kw);
  interp3_kernel<<<nblk((size_t)B * 1024 * S2, 256), 256, 0, stream>>>(
      mlo, ki, kw, fpin, 1024, S2, S2, 1536, B);
  copy_channels_kernel<<<nblk((size_t)B * 512 * S2, 256), 256, 0, stream>>>(
      sa2o, fpin, 512, 1024, 1536, S2, B);
  const float* f0 = run_mlp(fp0L, 2, fpin, B, S2, actA, actB, stats, stream);    // 256ch@S2

  // -------- FP stage 1: query=xyz1(S1), key=xyz2(S2), 256+320 -> (256,128)
  knn3_kernel<<<nblk((size_t)B * S1, 256), 256, 0, stream>>>(xyz1, xyz2, S1, S2, B, ki, kw);
  interp3_kernel<<<nblk((size_t)B * 256 * S1, 256), 256, 0, stream>>>(
      f0, ki, kw, fpin, 256, S2, S1, 576, B);
  copy_channels_kernel<<<nblk((size_t)B * 320 * S1, 256), 256, 0, stream>>>(
      sa1o, fpin, 320, 256, 576, S1, B);
  const float* f1 = run_mlp(fp1L, 2, fpin, B, S1, actA, actB, stats, stream);    // 128ch@S1

  // -------- FP stage 2: query=points(N0), key=xyz1(S1), 128+19 -> (128,128)
  knn3_kernel<<<nblk((size_t)B * N0, 256), 256, 0, stream>>>(points, xyz1, N0, S1, B, ki, kw);
  interp3_kernel<<<nblk((size_t)B * 128 * N0, 256), 256, 0, stream>>>(
      f1, ki, kw, fpin, 128, S1, N0, 147, B);
  copy_channels_kernel<<<nblk((size_t)B * 19 * N0, 256), 256, 0, stream>>>(
      feat0, fpin, 19, 128, 147, N0, B);
  const float* f2 = run_mlp(fp2L, 2, fpin, B, N0, actA, actB, stats, stream);    // 128ch@N0

  // -------- seg head: mlp_seg then 50x128 GEMM + bias
  const float* sfeat = run_mlp(segL, 1, f2, B, N0, actA, actB, stats, stream);   // 128ch@N0
  launch_gemm(seg_w, sfeat, (float*)d_out, 50, 128, N0, B, stream);
  const size_t ot = (size_t)B * 50 * N0;
  bias_add_kernel<<<nblk(ot, 256), 256, 0, stream>>>((float*)d_out, seg_b, 50, N0, ot);
}